// MLSMPO_85581518340709
// MI455X (gfx1250) — compile-verified
//
#include <hip/hip_runtime.h>

// ---------------------------------------------------------------------------
// MI455X / gfx1250 implementation of the MPS/MPO reference.
// Hot loops are chains of 16x16x16 fp32 matmuls -> V_WMMA_F32_16X16X4_F32
// (4 K-chunks per matmul, fp32 C accumulator preserves the 31-deep chain).
// Recurrent operand is re-laid-out D->A/B each step via a per-wave LDS
// round trip ordered with the CDNA5 split counter wait `s_wait_dscnt`.
// ---------------------------------------------------------------------------

typedef __attribute__((ext_vector_type(2))) float v2f;
typedef __attribute__((ext_vector_type(8))) float v8f;

#define LDS_FENCE() asm volatile("s_wait_dscnt 0x0" ::: "memory")

static __device__ __forceinline__ v8f wmma4(v2f a, v2f b, v8f c) {
  // D(16x16,f32) = A(16x4,f32) * B(4x16,f32) + C
  return __builtin_amdgcn_wmma_f32_16x16x4_f32(false, a, false, b,
                                               (short)0, c, false, false);
}

// ============================================================================
// Kernel 0: weight prep (input independent).
//   Wo2[l,f,v]  = sum_k W2_out[k,l,f,v]                  (640 floats)
//   inv_norm    = rsqrt( sum(c) ) after the 63-step c scan
// ============================================================================
__global__ void __launch_bounds__(640)
prep_kernel(const float* __restrict__ W2_out, const float* __restrict__ W2_mid,
            float* __restrict__ Wo2, float* __restrict__ inv_norm)
{
  __shared__ float wo[640];
  __shared__ float c[256], cn[256], t[4][256];
  const int tid = threadIdx.x;

  if (tid < 640) {                       // W2_out strides: k:640, l:40, f:10, v:1
    float s = 0.f;
    for (int k = 0; k < 16; ++k) s += W2_out[k * 640 + tid];
    wo[tid] = s;
    Wo2[tid] = s;
  }
  __syncthreads();

  if (tid < 256) {                       // c0[l,L] = sum_{f,v} Wo2[l,f,v]*Wo2[L,f,v]
    const int l = tid >> 4, L = tid & 15;
    float s = 0.f;
    for (int fv = 0; fv < 40; ++fv) s += wo[l * 40 + fv] * wo[L * 40 + fv];
    c[tid] = s;
  }
  __syncthreads();

  for (int s = 0; s < 63; ++s) {
    const float* Ws = W2_mid + s * 1024;         // [k*16+l]*4 + f
    if (tid < 256) {                             // t[f][k,L] = sum_K c[k,K]*W[K,L,f]
      const int k = tid >> 4, L = tid & 15;
      for (int f = 0; f < 4; ++f) {
        float acc = 0.f;
        for (int K = 0; K < 16; ++K) acc += c[k * 16 + K] * Ws[(K * 16 + L) * 4 + f];
        t[f][tid] = acc;
      }
    }
    __syncthreads();
    if (tid < 256) {                             // c'[l,L] = sum_{k,f} W[k,l,f]*t[f][k,L]
      const int l = tid >> 4, L = tid & 15;
      float acc = 0.f;
      for (int k = 0; k < 16; ++k) {
        const float* wp = Ws + (k * 16 + l) * 4;
        for (int f = 0; f < 4; ++f) acc += wp[f] * t[f][k * 16 + L];
      }
      cn[tid] = acc;
    }
    __syncthreads();
    if (tid < 256) c[tid] = cn[tid];
    __syncthreads();
  }

  if (tid == 0) {
    float s = 0.f;
    for (int e = 0; e < 256; ++e) s += c[e];
    inv_norm[0] = rsqrtf(s);
  }
}

// ============================================================================
// Kernel 1: layer 1.  grid = B blocks of 256 threads (8 waves = 8 MPO blocks).
// Each wave: site-normalize its 32 sites, build M_s in WMMA register layout,
// run the 31-matmul fp32 WMMA scan, produce T_blk (2 matrices) into LDS.
// Then the 256 threads run the Kronecker acc chain and normalize y.
// ============================================================================
__global__ void __launch_bounds__(256)
l1_kernel(const float* __restrict__ x,       // (B,32,32)
          const float* __restrict__ W1_out,  // (8,16,16,4,2)
          const float* __restrict__ W1_mid,  // (8,31,16,16,4)
          float* __restrict__ Y)             // (B,256)
{
  __shared__ float sP[8][256];        // per-wave D->A/B transpose scratch
  __shared__ float sT[8][2][256];     // T_blk[n][v][k*16+m]
  __shared__ float sAcc[2][16 * 256]; // acc ping-pong [k*256+u]
  __shared__ float sRed[8];

  const int b    = blockIdx.x;
  const int tid  = threadIdx.x;
  const int wv   = tid >> 5;          // wave == MPO block index n
  const int lane = tid & 31;
  const int n    = wv;
  const int hi   = lane >> 4;
  const int lo   = lane & 15;

  // ---- squeeze(k=2) + site-normalize; lane handles site s = lane of block n
  const int p  = n * 32 + lane;
  const int si = p >> 4, sj = p & 15;
  const float* xb = x + (size_t)b * 1024;
  float f0 = xb[(2 * si) * 32 + 2 * sj];
  float f1 = xb[(2 * si) * 32 + 2 * sj + 1];
  float f2 = xb[(2 * si + 1) * 32 + 2 * sj];
  float f3 = xb[(2 * si + 1) * 32 + 2 * sj + 1];
  {
    float rn = rsqrtf(f0 * f0 + f1 * f1 + f2 * f2 + f3 * f3);
    f0 *= rn; f1 *= rn; f2 *= rn; f3 *= rn;
  }

  const float* Wm = W1_mid + (size_t)n * 31 * 1024;

  // ---- P = M_1, built directly in WMMA A layout:
  //      chunk c, elem v: row r=lo, col k = 4c+v+2*hi
  v2f A[4];
  {
    float g0 = __shfl(f0, 1), g1 = __shfl(f1, 1);
    float g2 = __shfl(f2, 1), g3 = __shfl(f3, 1);
#pragma unroll
    for (int cc = 0; cc < 4; ++cc)
#pragma unroll
      for (int v = 0; v < 2; ++v) {
        int k = 4 * cc + v + 2 * hi;
        float4 w = *(const float4*)(Wm + (lo * 16 + k) * 4);
        A[cc][v] = g0 * w.x + g1 * w.y + g2 * w.z + g3 * w.w;
      }
  }

  // ---- scan: P <- P @ M_s for s = 2..31  (4 WMMAs per matmul)
  float* Pl = sP[wv];
#pragma unroll 1
  for (int s = 2; s <= 31; ++s) {
    const float* Ws = Wm + (s - 1) * 1024;
    float g0 = __shfl(f0, s), g1 = __shfl(f1, s);
    float g2 = __shfl(f2, s), g3 = __shfl(f3, s);
    v2f Bm[4];                        // M_s in B layout: elem (k=4c+v+2hi, col=lo)
#pragma unroll
    for (int cc = 0; cc < 4; ++cc)
#pragma unroll
      for (int v = 0; v < 2; ++v) {
        int k = 4 * cc + v + 2 * hi;
        float4 w = *(const float4*)(Ws + (k * 16 + lo) * 4);
        Bm[cc][v] = g0 * w.x + g1 * w.y + g2 * w.z + g3 * w.w;
      }
    v8f D = {};
#pragma unroll
    for (int cc = 0; cc < 4; ++cc) D = wmma4(A[cc], Bm[cc], D);

    // D layout -> plain [m*16+col] in LDS, then reload in A layout
#pragma unroll
    for (int r = 0; r < 8; ++r) Pl[(r + 8 * hi) * 16 + lo] = D[r];
    LDS_FENCE();
#pragma unroll
    for (int cc = 0; cc < 4; ++cc) {
      A[cc][0] = Pl[lo * 16 + 4 * cc + 2 * hi];
      A[cc][1] = Pl[lo * 16 + 4 * cc + 2 * hi + 1];
    }
  }

  // ---- T_blk[:,:,v] = T_out[:,:,v] @ P   (P read back in B layout)
  LDS_FENCE();
  v2f Bp[4];
#pragma unroll
  for (int cc = 0; cc < 4; ++cc) {
    Bp[cc][0] = Pl[(4 * cc + 0 + 2 * hi) * 16 + lo];
    Bp[cc][1] = Pl[(4 * cc + 1 + 2 * hi) * 16 + lo];
  }
  {
    float o0 = __shfl(f0, 0), o1 = __shfl(f1, 0);
    float o2 = __shfl(f2, 0), o3 = __shfl(f3, 0);
    const float* Wo = W1_out + (size_t)n * 2048;  // [k*16+l]*8 + f*2 + v
    v2f A0[4], A1[4];
#pragma unroll
    for (int cc = 0; cc < 4; ++cc)
#pragma unroll
      for (int u = 0; u < 2; ++u) {
        int l = 4 * cc + u + 2 * hi;
        const float* wp = Wo + (lo * 16 + l) * 8;
        float4 wa = *(const float4*)(wp);       // f0v0 f0v1 f1v0 f1v1
        float4 wb = *(const float4*)(wp + 4);   // f2v0 f2v1 f3v0 f3v1
        A0[cc][u] = o0 * wa.x + o1 * wa.z + o2 * wb.x + o3 * wb.z;
        A1[cc][u] = o0 * wa.y + o1 * wa.w + o2 * wb.y + o3 * wb.w;
      }
    v8f D0 = {}, D1 = {};
#pragma unroll
    for (int cc = 0; cc < 4; ++cc) {
      D0 = wmma4(A0[cc], Bp[cc], D0);
      D1 = wmma4(A1[cc], Bp[cc], D1);
    }
#pragma unroll
    for (int r = 0; r < 8; ++r) {
      sT[n][0][(r + 8 * hi) * 16 + lo] = D0[r];
      sT[n][1][(r + 8 * hi) * 16 + lo] = D1[r];
    }
  }
  if (tid < 16) sAcc[0][tid * 256] = 1.0f;   // acc0[k,0] = 1
  __syncthreads();

  // ---- Kronecker acc chain: acc'[m, 2u+v] = sum_k acc[k,u] * T[n][v][k,m]
  int cur = 0;
#pragma unroll 1
  for (int nb = 0; nb < 8; ++nb) {
    const int U  = 1 << nb;
    const int CO = 2 * U;
    const float* src = sAcc[cur];
    float* dst = sAcc[cur ^ 1];
    for (int e = tid; e < 16 * CO; e += 256) {
      const int m = e / CO, col = e - m * CO;
      const int u = col >> 1, vv = col & 1;
      float acc = 0.f;
      const float* tm = &sT[nb][vv][m];
#pragma unroll
      for (int k = 0; k < 16; ++k) acc += src[k * 256 + u] * tm[k * 16];
      dst[m * 256 + col] = acc;
    }
    __syncthreads();
    cur ^= 1;
  }

  // ---- y[j] = sum_m acc[m,j]; normalize over 256 and write out
  const float* fin = sAcc[cur];
  float yv = 0.f;
#pragma unroll
  for (int m = 0; m < 16; ++m) yv += fin[m * 256 + tid];
  float y2 = yv * yv;
#pragma unroll
  for (int off = 16; off > 0; off >>= 1) y2 += __shfl_down(y2, off);
  if (lane == 0) sRed[wv] = y2;
  __syncthreads();
  float tot = 0.f;
#pragma unroll
  for (int w = 0; w < 8; ++w) tot += sRed[w];
  Y[(size_t)b * 256 + tid] = yv * rsqrtf(tot);
}

// ============================================================================
// Kernel 2: batch-norm statistics over the batch axis.
// ============================================================================
__global__ void __launch_bounds__(1024)
bn_kernel(const float* __restrict__ Y, float* __restrict__ mu,
          float* __restrict__ var, int B)
{
  __shared__ float sSum[1024], sSq[1024];
  const int j = threadIdx.x & 255;
  const int grp = threadIdx.x >> 8;
  float s = 0.f, q = 0.f;
  for (int b = grp; b < B; b += 4) {
    float v = Y[(size_t)b * 256 + j];
    s += v; q += v * v;
  }
  sSum[threadIdx.x] = s; sSq[threadIdx.x] = q;
  __syncthreads();
  if (threadIdx.x < 256) {
    float ts = 0.f, tq = 0.f;
    for (int g = 0; g < 4; ++g) { ts += sSum[g * 256 + j]; tq += sSq[g * 256 + j]; }
    const float invB = 1.0f / (float)B;
    float m = ts * invB;
    mu[j] = m;
    var[j] = tq * invB - m * m;
  }
}

// ============================================================================
// Kernel 3: layer 2.  grid = B blocks of 32 threads (one wave per batch).
// BN apply -> squeeze -> site normalize -> init A -> 63-step WMMA scan.
// State S (16x16 D-regs, cols 10..15 zero); per step S <- mats_s^T @ S.
// ============================================================================
__global__ void __launch_bounds__(32)
l2_kernel(const float* __restrict__ Y, const float* __restrict__ mu,
          const float* __restrict__ var, const float* __restrict__ gamma,
          const float* __restrict__ beta, const float* __restrict__ W2_mid,
          const float* __restrict__ Wo2, const float* __restrict__ inv_norm,
          float* __restrict__ out)
{
  __shared__ float z[64 * 4];   // normalized site features
  __shared__ float sS[256];     // state D->B transpose scratch

  const int b = blockIdx.x;
  const int lane = threadIdx.x;
  const int hi = lane >> 4, lo = lane & 15;

  // ---- BN apply + squeeze(k=2) + site normalize (2 sites per lane)
  for (int p = lane; p < 64; p += 32) {
    const int i = p >> 3, j = p & 7;
    float zv[4];
#pragma unroll
    for (int di = 0; di < 2; ++di)
#pragma unroll
      for (int dj = 0; dj < 2; ++dj) {
        int idx = (2 * i + di) * 16 + (2 * j + dj);
        float yv = Y[(size_t)b * 256 + idx];
        zv[di * 2 + dj] =
            gamma[idx] * (yv - mu[idx]) * rsqrtf(var[idx] + 1e-5f) + beta[idx];
      }
    float rn = rsqrtf(zv[0] * zv[0] + zv[1] * zv[1] + zv[2] * zv[2] + zv[3] * zv[3]);
#pragma unroll
    for (int f = 0; f < 4; ++f) z[p * 4 + f] = zv[f] * rn;
  }
  __syncthreads();

  // ---- init S[m,v] = sum_f z0[f] * Wo2[m,f,v]   (D layout; v>=10 -> 0)
  v8f S;
#pragma unroll
  for (int r = 0; r < 8; ++r) {
    const int m = r + 8 * hi;
    float acc = 0.f;
    if (lo < 10) {
#pragma unroll
      for (int f = 0; f < 4; ++f) acc += z[f] * Wo2[(m * 4 + f) * 10 + lo];
    }
    S[r] = acc;
  }

  // ---- scan: S <- mats_s^T @ S for s = 0..62
#pragma unroll 1
  for (int s = 0; s < 63; ++s) {
    const float* Ws = W2_mid + s * 1024;
    const float g0 = z[(s + 1) * 4 + 0], g1 = z[(s + 1) * 4 + 1];
    const float g2 = z[(s + 1) * 4 + 2], g3 = z[(s + 1) * 4 + 3];
    v2f Am[4];                        // mats^T in A layout: elem mats[k, l=lo]
#pragma unroll
    for (int cc = 0; cc < 4; ++cc)
#pragma unroll
      for (int v = 0; v < 2; ++v) {
        int k = 4 * cc + v + 2 * hi;
        float4 w = *(const float4*)(Ws + (k * 16 + lo) * 4);
        Am[cc][v] = g0 * w.x + g1 * w.y + g2 * w.z + g3 * w.w;
      }
    // state D layout -> plain LDS -> B layout
#pragma unroll
    for (int r = 0; r < 8; ++r) sS[(r + 8 * hi) * 16 + lo] = S[r];
    LDS_FENCE();
    v2f Bs[4];
#pragma unroll
    for (int cc = 0; cc < 4; ++cc) {
      Bs[cc][0] = sS[(4 * cc + 0 + 2 * hi) * 16 + lo];
      Bs[cc][1] = sS[(4 * cc + 1 + 2 * hi) * 16 + lo];
    }
    v8f D = {};
#pragma unroll
    for (int cc = 0; cc < 4; ++cc) D = wmma4(Am[cc], Bs[cc], D);
    S = D;
  }

  // ---- out[b,v] = inv_norm * sum_m S[m,v]
  float part = 0.f;
#pragma unroll
  for (int r = 0; r < 8; ++r) part += S[r];
  part += __shfl_down(part, 16);     // rows 0..7 + rows 8..15
  if (lane < 10) out[(size_t)b * 10 + lane] = part * inv_norm[0];
}

// ============================================================================
extern "C" void kernel_launch(void* const* d_in, const int* in_sizes, int n_in,
                              void* d_out, int out_size, void* d_ws, size_t ws_size,
                              hipStream_t stream)
{
  const float* x      = (const float*)d_in[0];
  const float* W1_out = (const float*)d_in[1];
  const float* W1_mid = (const float*)d_in[2];
  const float* W2_out = (const float*)d_in[3];
  const float* W2_mid = (const float*)d_in[4];
  const float* gamma  = (const float*)d_in[5];
  const float* beta   = (const float*)d_in[6];
  float* out = (float*)d_out;

  const int B = in_sizes[0] / 1024;   // (B,32,32,1)

  float* ws   = (float*)d_ws;
  float* Yb   = ws;                   // B*256
  float* mu   = ws + (size_t)B * 256;
  float* var  = mu + 256;
  float* Wo2  = var + 256;            // 640
  float* invn = Wo2 + 640;            // 1

  hipLaunchKernelGGL(prep_kernel, dim3(1), dim3(640), 0, stream,
                     W2_out, W2_mid, Wo2, invn);
  hipLaunchKernelGGL(l1_kernel, dim3(B), dim3(256), 0, stream,
                     x, W1_out, W1_mid, Yb);
  hipLaunchKernelGGL(bn_kernel, dim3(1), dim3(1024), 0, stream, Yb, mu, var, B);
  hipLaunchKernelGGL(l2_kernel, dim3(B), dim3(32), 0, stream,
                     Yb, mu, var, gamma, beta, W2_mid, Wo2, invn, out);
}